// MMFNet_83425444757854
// MI455X (gfx1250) — compile-verified
//
#include <hip/hip_runtime.h>

// Problem constants from the reference
#define BN 1024   // batch
#define SN 127    // struct features
#define DN 512    // model dim
#define HN 8      // heads
#define TN 128    // tokens (S+1)
#define DHN 64    // head dim

typedef __attribute__((ext_vector_type(2))) float v2f;
typedef __attribute__((ext_vector_type(8))) float v8f;

// ---------------------------------------------------------------------------
// One 16x64 fp32 output tile per wave via V_WMMA_F32_16X16X4_F32.
// The A fragment (strided row loads) is loaded once per k-step and reused
// across 4 N-subtiles; B fragment loads are lane-contiguous (coalesced).
// Fragment layouts per CDNA5 ISA 7.12.2:
//   A 16x4 : lane<16 -> (A[m][k],A[m][k+1]) ; lane>=16 -> (A[m][k+2],A[m][k+3])
//   B 4x16 : VGPR0 = rows k / k+2 striped across lane halves, VGPR1 = k+1 / k+3
//   C/D    : VGPR j -> row (8*half + j), col = lane&15
// ---------------------------------------------------------------------------
__device__ __forceinline__ void wmma_tile16x64_f32(
    const float* __restrict__ A, int lda,
    const float* __restrict__ B, int ldb,
    float* __restrict__ C, int ldc, int K)
{
    const int lane = threadIdx.x & 31;
    const int half = lane >> 4;   // 0: lanes 0-15, 1: lanes 16-31
    const int lr   = lane & 15;

    v8f acc0 = {0.f,0.f,0.f,0.f,0.f,0.f,0.f,0.f};
    v8f acc1 = acc0, acc2 = acc0, acc3 = acc0;

    for (int k = 0; k < K; k += 4) {
        v2f a;
        const float* ap = A + lr * lda + k + 2 * half;
        a.x = ap[0];
        a.y = ap[1];

        const float* bp = B + (k + 2 * half) * ldb + lr;
        v2f b0, b1, b2, b3;
        b0.x = bp[0];        b0.y = bp[ldb];
        b1.x = bp[16];       b1.y = bp[ldb + 16];
        b2.x = bp[32];       b2.y = bp[ldb + 32];
        b3.x = bp[48];       b3.y = bp[ldb + 48];

        acc0 = __builtin_amdgcn_wmma_f32_16x16x4_f32(false, a, false, b0,
                                                     (short)0, acc0, false, false);
        acc1 = __builtin_amdgcn_wmma_f32_16x16x4_f32(false, a, false, b1,
                                                     (short)0, acc1, false, false);
        acc2 = __builtin_amdgcn_wmma_f32_16x16x4_f32(false, a, false, b2,
                                                     (short)0, acc2, false, false);
        acc3 = __builtin_amdgcn_wmma_f32_16x16x4_f32(false, a, false, b3,
                                                     (short)0, acc3, false, false);
    }

    float* cp = C + (8 * half) * ldc + lr;
#pragma unroll
    for (int j = 0; j < 8; ++j) {
        cp[j * ldc +  0] = acc0[j];
        cp[j * ldc + 16] = acc1[j];
        cp[j * ldc + 32] = acc2[j];
        cp[j * ldc + 48] = acc3[j];
    }
}

// ---------------------------------------------------------------------------
// Kernel 1: yq/yk/yv = X0 @ {Wq,Wk,Wv}.
// grid = (N/64, M/64, 3), block = 128 (4 waves; wave w owns M-subtile w).
// ---------------------------------------------------------------------------
__global__ void proj_gemm_kernel(const float* __restrict__ X0,
                                 const float* __restrict__ Wq,
                                 const float* __restrict__ Wk,
                                 const float* __restrict__ Wv,
                                 float* __restrict__ yq,
                                 float* __restrict__ yk,
                                 float* __restrict__ yv)
{
    const float* W;
    float* Y;
    if (blockIdx.z == 0)      { W = Wq; Y = yq; }
    else if (blockIdx.z == 1) { W = Wk; Y = yk; }
    else                      { W = Wv; Y = yv; }
    const int wave = threadIdx.x >> 5;
    const int n0 = blockIdx.x * 64;
    const int m0 = (blockIdx.y * 4 + wave) * 16;
    wmma_tile16x64_f32(X0 + (size_t)m0 * DN, DN,
                       W + n0, DN,
                       Y + (size_t)m0 * DN + n0, DN, DN);
}

// ---------------------------------------------------------------------------
// Kernel 2: c[b,h] = scale * dot(yq[b,h,:], yk[b,h,:]) over DH=64.
// ---------------------------------------------------------------------------
__global__ void head_dot_kernel(const float* __restrict__ yq,
                                const float* __restrict__ yk,
                                float* __restrict__ c)
{
    const int idx = blockIdx.x * blockDim.x + threadIdx.x;  // BN*HN = 8192
    if (idx >= BN * HN) return;
    const int b = idx >> 3;
    const int h = idx & 7;
    const float* q = yq + (size_t)b * DN + h * DHN;
    const float* k = yk + (size_t)b * DN + h * DHN;
    float s = 0.f;
#pragma unroll 8
    for (int j = 0; j < DHN; ++j) s += q[j] * k[j];
    c[idx] = s * 0.125f;  // 1/sqrt(64)
}

// ---------------------------------------------------------------------------
// Kernel 3: z[b,h,:] = yv[b, h*64:(h+1)*64] @ Wo[h*64:(h+1)*64, :]
// 8 independent [1024x64]x[64x512] GEMMs.
// grid = (N/64, M/64, H), block = 128.  z layout [b][h][d]: row stride H*D.
// ---------------------------------------------------------------------------
__global__ void zhead_gemm_kernel(const float* __restrict__ yv,
                                  const float* __restrict__ Wo,
                                  float* __restrict__ z)
{
    const int h  = blockIdx.z;
    const int wave = threadIdx.x >> 5;
    const int n0 = blockIdx.x * 64;
    const int m0 = (blockIdx.y * 4 + wave) * 16;
    wmma_tile16x64_f32(yv + (size_t)m0 * DN + h * DHN, DN,
                       Wo + (size_t)(h * DHN) * DN + n0, DN,
                       z + (size_t)m0 * (HN * DN) + h * DN + n0, HN * DN, DHN);
}

// ---------------------------------------------------------------------------
// Kernel 4: fused rank-1 softmax weights + rank-8 output combine.
// logits_u = (s_t * c[b,h]) * s_u ;  w[b,t,h] = sum_u softmax_u * s_u
// out[b,t,d] = sum_h w[b,t,h] * z[b,h,d]  (256MB bandwidth-bound store)
// grid = (B, T/32), block = 256 (8 waves; wave w owns head h=w).
// ---------------------------------------------------------------------------
__global__ void attn_out_kernel(const float* __restrict__ Xs,   // [B,S]
                                const float* __restrict__ cbh,  // [B,H]
                                const float* __restrict__ z,    // [B,H,D]
                                float* __restrict__ out)        // [B,T,D]
{
    __shared__ float s_s[TN];        // token scalars s_u
    __shared__ float z_s[HN * DN];   // 16KB z tile for this batch
    __shared__ float w_s[HN];

    const int b    = blockIdx.x;
    const int t0   = blockIdx.y * 32;
    const int tid  = threadIdx.x;
    const int h    = tid >> 5;       // wave index == head
    const int lane = tid & 31;

    if (tid < TN) s_s[tid] = (tid == 0) ? 1.0f : Xs[(size_t)b * SN + tid - 1];
    for (int i = tid; i < HN * DN; i += 256) z_s[i] = z[(size_t)b * HN * DN + i];
    __syncthreads();

    const float ch  = cbh[b * HN + h];
    const float su0 = s_s[lane], su1 = s_s[lane + 32],
                su2 = s_s[lane + 64], su3 = s_s[lane + 96];

    for (int tt = 0; tt < 32; ++tt) {
        const int t = t0 + tt;
        const float a = ch * s_s[t];

        // logits (each lane covers 4 key positions)
        float l0 = a * su0, l1 = a * su1, l2 = a * su2, l3 = a * su3;
        float m = fmaxf(fmaxf(l0, l1), fmaxf(l2, l3));
#pragma unroll
        for (int off = 16; off > 0; off >>= 1)
            m = fmaxf(m, __shfl_xor(m, off, 32));

        const float e0 = __expf(l0 - m), e1 = __expf(l1 - m),
                    e2 = __expf(l2 - m), e3 = __expf(l3 - m);
        float den = e0 + e1 + e2 + e3;
        float num = e0 * su0 + e1 * su1 + e2 * su2 + e3 * su3;
#pragma unroll
        for (int off = 16; off > 0; off >>= 1) {
            den += __shfl_xor(den, off, 32);
            num += __shfl_xor(num, off, 32);
        }
        if (lane == 0) w_s[h] = num / den;
        __syncthreads();

        // 512 outputs per token: thread owns d = 2*tid (float2 store)
        const int d = tid * 2;
        float r0 = 0.f, r1 = 0.f;
#pragma unroll
        for (int hh = 0; hh < HN; ++hh) {
            const float w = w_s[hh];
            r0 += w * z_s[hh * DN + d];
            r1 += w * z_s[hh * DN + d + 1];
        }
        float2* op = (float2*)(out + ((size_t)(b * TN + t)) * DN + d);
        *op = make_float2(r0, r1);
        __syncthreads();  // protect w_s for next token
    }
}

// ---------------------------------------------------------------------------
// Launch
// ---------------------------------------------------------------------------
extern "C" void kernel_launch(void* const* d_in, const int* in_sizes, int n_in,
                              void* d_out, int out_size, void* d_ws, size_t ws_size,
                              hipStream_t stream)
{
    const float* X0 = (const float*)d_in[0];  // [B,D]
    const float* Xs = (const float*)d_in[1];  // [B,S]
    const float* Wq = (const float*)d_in[2];  // [D,D]
    const float* Wk = (const float*)d_in[3];
    const float* Wv = (const float*)d_in[4];
    const float* Wo = (const float*)d_in[5];
    float* out = (float*)d_out;               // [B,T,D]

    // Workspace layout (floats); total ~23.1 MB
    float* ws  = (float*)d_ws;
    float* yq  = ws;                          // B*D
    float* yk  = yq + (size_t)BN * DN;        // B*D
    float* yv  = yk + (size_t)BN * DN;        // B*D
    float* cbh = yv + (size_t)BN * DN;        // B*H
    float* z   = cbh + (size_t)BN * HN;       // B*H*D

    // 1) QKV projections: 3x [1024x512]x[512x512] fp32 WMMA GEMMs
    proj_gemm_kernel<<<dim3(DN / 64, BN / 64, 3), 128, 0, stream>>>(
        X0, Wq, Wk, Wv, yq, yk, yv);

    // 2) per-(b,h) scaled q.k dot products
    head_dot_kernel<<<(BN * HN + 255) / 256, 256, 0, stream>>>(yq, yk, cbh);

    // 3) per-head z = yv_h @ Wo_h : 8x [1024x64]x[64x512] fp32 WMMA GEMMs
    zhead_gemm_kernel<<<dim3(DN / 64, BN / 64, HN), 128, 0, stream>>>(yv, Wo, z);

    // 4) fused rank-1 softmax + rank-8 combine (bandwidth-bound 256MB store)
    attn_out_kernel<<<dim3(BN, TN / 32), 256, 0, stream>>>(Xs, cbh, z, out);
}